// GAT_MUTAG_59725815218464
// MI455X (gfx1250) — compile-verified
//
#include <hip/hip_runtime.h>
#include <hip/hip_bf16.h>

#define DH 32
#define F_IN 7
#define F_E 4
#define SLOPE 0.2f
#define EPS 1e-5f

typedef __attribute__((ext_vector_type(2))) float v2f;
typedef __attribute__((ext_vector_type(8))) float v8f;

// ---------- helpers ----------
__device__ __forceinline__ unsigned fkey(float f) {
    unsigned b = __float_as_uint(f);
    return (b & 0x80000000u) ? ~b : (b | 0x80000000u);
}
__device__ __forceinline__ float funkey(unsigned k) {
    unsigned b = (k & 0x80000000u) ? (k & 0x7FFFFFFFu) : ~k;
    return __uint_as_float(b);
}

// ---------- tiny precompute: weA[l][f] = sum_d We[l][f][d] * att_edge[l][d] ----------
__global__ void prep_weA_kernel(const float* __restrict__ We,
                                const float* __restrict__ att_edge,
                                float* __restrict__ weA) {
    int i = threadIdx.x;
    if (i >= 3 * F_E) return;
    int l = i >> 2, f = i & 3;
    float s = 0.f;
    for (int d = 0; d < DH; ++d)
        s += We[(l * F_E + f) * DH + d] * att_edge[l * DH + d];
    weA[i] = s;
}

// ---------- layer 0 projection: h = x[N,7] @ W0[7,32] (scalar, K too small for WMMA) ----------
__global__ __launch_bounds__(256) void proj0_kernel(const float* __restrict__ x,
                                                    const float* __restrict__ W0,
                                                    float* __restrict__ h, int n) {
    int i = blockIdx.x * blockDim.x + threadIdx.x;
    if (i >= n * DH) return;
    int node = i >> 5, c = i & 31;
    float acc = 0.f;
#pragma unroll
    for (int f = 0; f < F_IN; ++f)
        acc = fmaf(x[node * F_IN + f], W0[f * DH + c], acc);
    h[i] = acc;
}

// ---------- layers 1-2 projection via V_WMMA_F32_16X16X4_F32 ----------
// h_out[N,32] = h_in[N,32] @ W[32,32]; one wave -> one 16x16 tile, 8 chained K=4 WMMAs.
__global__ __launch_bounds__(256) void proj_wmma_kernel(const float* __restrict__ hin,
                                                        const float* __restrict__ W,
                                                        float* __restrict__ hout,
                                                        int nrows) {
    int wave = (blockIdx.x * blockDim.x + threadIdx.x) >> 5;
    int lane = threadIdx.x & 31;
    int rowTiles = nrows >> 4;                  // nrows is a multiple of 16 (100000)
    int rowTile = wave >> 1;
    int colTile = wave & 1;
    if (rowTile >= rowTiles) return;            // wave-uniform exit: EXEC stays all-ones
    int m0 = rowTile << 4;
    int n0 = colTile << 4;
    int half = lane >> 4;                       // 0: lanes0-15, 1: lanes16-31
    int l16 = lane & 15;

    const float* arow = hin + (size_t)(m0 + l16) * DH + half * 2;   // A: M=l16, K base
    int bcol = n0 + l16;

    v8f acc = {};
#pragma unroll
    for (int j = 0; j < 8; ++j) {
        // A 16x4 f32: v0 = K(4j + 2*half), v1 = K(4j + 2*half + 1)
        v2f a = *(const v2f*)(arow + 4 * j);
        // B 4x16 f32: v0 = row K(4j + 2*half), v1 = row K(4j + 2*half + 1), lane = N
        int kr = 4 * j + 2 * half;
        v2f b;
        b.x = W[kr * DH + bcol];
        b.y = W[(kr + 1) * DH + bcol];
        acc = __builtin_amdgcn_wmma_f32_16x16x4_f32(
            /*neg_a=*/false, a, /*neg_b=*/false, b,
            /*c_mod=*/(short)0, acc, /*reuse_a=*/false, /*reuse_b=*/false);
    }
    // D layout: VGPR v -> M = m0 + v + 8*half, N = n0 + l16
#pragma unroll
    for (int v = 0; v < 8; ++v)
        hout[(size_t)(m0 + v + 8 * half) * DH + n0 + l16] = acc[v];
}

// ---------- per-node attention scalars: alpha_src = h . a_s, alpha_dst = h . a_d ----------
__global__ __launch_bounds__(256) void node_alpha_kernel(const float* __restrict__ h,
                                                         const float* __restrict__ a_s,
                                                         const float* __restrict__ a_d,
                                                         float* __restrict__ alpha_src,
                                                         float* __restrict__ alpha_dst,
                                                         int n) {
    int i = blockIdx.x * blockDim.x + threadIdx.x;
    int node = i >> 5, c = i & 31;              // one wave32 == one node's 32 channels
    if (node >= n) return;                      // wave-uniform
    float v = h[(size_t)node * DH + c];
    float s = v * a_s[c];
    float d = v * a_d[c];
#pragma unroll
    for (int m = 16; m >= 1; m >>= 1) {
        s += __shfl_xor(s, m, 32);
        d += __shfl_xor(d, m, 32);
    }
    if (c == 0) { alpha_src[node] = s; alpha_dst[node] = d; }
}

// ---------- per-layer init: aggr = bias, maxkey = 0, den = 0, sums = 0 ----------
__global__ __launch_bounds__(256) void layer_init_kernel(unsigned* __restrict__ maxkey,
                                                         float* __restrict__ den,
                                                         float* __restrict__ aggr,
                                                         const float* __restrict__ bias,
                                                         float* __restrict__ sums, int n) {
    int i = blockIdx.x * blockDim.x + threadIdx.x;
    if (i < n * DH) aggr[i] = bias[i & 31];
    if (i < n) { maxkey[i] = 0u; den[i] = 0.f; }
    if (i < 2 * DH) sums[i] = 0.f;
}

// ---------- edge pass 1: logits + segment max (ordered-uint atomicMax) ----------
__global__ __launch_bounds__(256) void edge_logits_kernel(const int* __restrict__ ei,
                                                          const float* __restrict__ ea,
                                                          const float* __restrict__ asrc,
                                                          const float* __restrict__ adst,
                                                          const float* __restrict__ weA,
                                                          float* __restrict__ elog,
                                                          unsigned* __restrict__ maxkey,
                                                          int E_) {
    int e = blockIdx.x * blockDim.x + threadIdx.x;
    if (e >= E_) return;
    int s = ei[e], d = ei[E_ + e];
    const float* a = ea + (size_t)e * F_E;
    float ae = a[0] * weA[0] + a[1] * weA[1] + a[2] * weA[2] + a[3] * weA[3];
    float x = asrc[s] + adst[d] + ae;
    float lg = x > 0.f ? x : SLOPE * x;         // leaky relu
    elog[e] = lg;
    atomicMax(&maxkey[d], fkey(lg));
}

// ---------- edge pass 2: num = exp(logit - m[dst]); den[dst] += num ----------
__global__ __launch_bounds__(256) void edge_exp_kernel(const int* __restrict__ ei,
                                                       float* __restrict__ elog,
                                                       const unsigned* __restrict__ maxkey,
                                                       float* __restrict__ den, int E_) {
    int e = blockIdx.x * blockDim.x + threadIdx.x;
    if (e >= E_) return;
    int d = ei[E_ + e];
    float num = __expf(elog[e] - funkey(maxkey[d]));
    elog[e] = num;
    atomicAdd(&den[d], num);
}

// ---------- edge pass 3: aggr[dst] += h[src] * (num/den[dst]) ----------
__global__ __launch_bounds__(256) void edge_aggr_kernel(const int* __restrict__ ei,
                                                        const float* __restrict__ num,
                                                        const float* __restrict__ den,
                                                        const float* __restrict__ h,
                                                        float* __restrict__ aggr, int E_) {
    int gid = blockIdx.x * blockDim.x + threadIdx.x;
    if (gid >= E_ * DH) return;
    int e = gid >> 5, c = gid & 31;
    int s = ei[e], d = ei[E_ + e];
    float alpha = num[e] / den[d];
    atomicAdd(&aggr[(size_t)d * DH + c], h[(size_t)s * DH + c] * alpha);
}

// ---------- BN stats: per-channel sum / sumsq over all N nodes ----------
__global__ __launch_bounds__(256) void bn_stats_kernel(const float* __restrict__ aggr,
                                                       float* __restrict__ sums, int n) {
    __shared__ float sv[256], sq[256];
    int i = blockIdx.x * 256 + threadIdx.x;
    float v = (i < n * DH) ? aggr[i] : 0.f;
    sv[threadIdx.x] = v;
    sq[threadIdx.x] = v * v;
    __syncthreads();
    if (threadIdx.x < DH) {                     // thread c reduces channel c within block
        float a = 0.f, b = 0.f;
#pragma unroll
        for (int k = 0; k < 256; k += DH) { a += sv[k + threadIdx.x]; b += sq[k + threadIdx.x]; }
        atomicAdd(&sums[threadIdx.x], a);
        atomicAdd(&sums[DH + threadIdx.x], b);
    }
}

// ---------- BN apply + ReLU (in place) ----------
__global__ __launch_bounds__(256) void bn_apply_kernel(float* __restrict__ aggr,
                                                       const float* __restrict__ sums,
                                                       const float* __restrict__ gamma,
                                                       const float* __restrict__ beta,
                                                       int n) {
    int i = blockIdx.x * blockDim.x + threadIdx.x;
    if (i >= n * DH) return;
    int c = i & 31;
    float inv_n = 1.f / (float)n;
    float mean = sums[c] * inv_n;
    float var = sums[DH + c] * inv_n - mean * mean;
    float sc = gamma[c] * rsqrtf(var + EPS);
    float v = (aggr[i] - mean) * sc + beta[c];
    aggr[i] = v > 0.f ? v : 0.f;
}

// ---------- pooling ----------
__global__ __launch_bounds__(256) void zero_kernel(float* __restrict__ p, int n) {
    int i = blockIdx.x * blockDim.x + threadIdx.x;
    if (i < n) p[i] = 0.f;
}
__global__ __launch_bounds__(256) void pool_kernel(const float* __restrict__ h,
                                                   const int* __restrict__ batch,
                                                   float* __restrict__ gemb, int n) {
    int i = blockIdx.x * blockDim.x + threadIdx.x;
    if (i >= n * DH) return;
    int node = i >> 5, c = i & 31;
    atomicAdd(&gemb[(size_t)batch[node] * DH + c], h[i]);
}

// ---------- head: logits = gemb @ fcW + fcb, 2-way softmax ----------
__global__ __launch_bounds__(256) void head_kernel(const float* __restrict__ gemb,
                                                   const float* __restrict__ fcW,
                                                   const float* __restrict__ fcb,
                                                   float* __restrict__ out, int nb) {
    int g = blockIdx.x * blockDim.x + threadIdx.x;
    if (g >= nb) return;
    float l0 = fcb[0], l1 = fcb[1];
#pragma unroll
    for (int c = 0; c < DH; ++c) {
        float v = gemb[(size_t)g * DH + c];
        l0 = fmaf(v, fcW[c * 2 + 0], l0);
        l1 = fmaf(v, fcW[c * 2 + 1], l1);
    }
    float m = fmaxf(l0, l1);
    float e0 = __expf(l0 - m), e1 = __expf(l1 - m);
    float s = 1.f / (e0 + e1);
    out[g * 2 + 0] = e0 * s;
    out[g * 2 + 1] = e1 * s;
}

extern "C" void kernel_launch(void* const* d_in, const int* in_sizes, int n_in,
                              void* d_out, int out_size, void* d_ws, size_t ws_size,
                              hipStream_t stream) {
    const float* x        = (const float*)d_in[0];
    const int*   ei       = (const int*)d_in[1];
    const int*   batch    = (const int*)d_in[2];
    const float* ea       = (const float*)d_in[3];
    const float* W0       = (const float*)d_in[4];
    const float* W12      = (const float*)d_in[5];
    const float* We       = (const float*)d_in[6];
    const float* att_src  = (const float*)d_in[7];
    const float* att_dst  = (const float*)d_in[8];
    const float* att_edge = (const float*)d_in[9];
    const float* bias     = (const float*)d_in[10];
    const float* gamma    = (const float*)d_in[11];
    const float* beta     = (const float*)d_in[12];
    const float* fcW      = (const float*)d_in[13];
    const float* fcb      = (const float*)d_in[14];
    float* out = (float*)d_out;

    const int N  = in_sizes[2];        // 100000
    const int E  = in_sizes[1] / 2;    // 3200000
    const int NB = out_size / 2;       // 512

    // workspace carve-out
    size_t off = 0;
    auto carve = [&](size_t bytes) {
        void* p = (char*)d_ws + off;
        off = (off + bytes + 255) & ~(size_t)255;
        return p;
    };
    float*    bufA   = (float*)carve((size_t)N * DH * 4);   // projected h
    float*    bufB   = (float*)carve((size_t)N * DH * 4);   // aggr / layer output
    float*    asrc   = (float*)carve((size_t)N * 4);
    float*    adst   = (float*)carve((size_t)N * 4);
    unsigned* maxkey = (unsigned*)carve((size_t)N * 4);
    float*    den    = (float*)carve((size_t)N * 4);
    float*    elog   = (float*)carve((size_t)E * 4);        // logits -> exp(num)
    float*    sums   = (float*)carve(2 * DH * 4);
    float*    weA    = (float*)carve(3 * F_E * 4);
    float*    gemb   = (float*)carve((size_t)NB * DH * 4);
    (void)ws_size; (void)n_in;

    const int TB = 256;
    int gN32 = (N * DH + TB - 1) / TB;          // N*32 threads
    int gE   = (E + TB - 1) / TB;               // E threads
    int gE32 = (int)(((long long)E * DH + TB - 1) / TB);
    int gWmma = ((N >> 4) * 2 + (TB / 32) - 1) / (TB / 32);  // waves->blocks

    prep_weA_kernel<<<1, 32, 0, stream>>>(We, att_edge, weA);

    for (int l = 0; l < 3; ++l) {
        // 1) projection
        if (l == 0)
            proj0_kernel<<<gN32, TB, 0, stream>>>(x, W0, bufA, N);
        else
            proj_wmma_kernel<<<gWmma, TB, 0, stream>>>(bufB, W12 + (size_t)(l - 1) * DH * DH,
                                                       bufA, N);
        // 2) per-node attention scalars
        node_alpha_kernel<<<gN32, TB, 0, stream>>>(bufA, att_src + l * DH, att_dst + l * DH,
                                                   asrc, adst, N);
        // 3) init aggr/bias, max, den, sums
        layer_init_kernel<<<gN32, TB, 0, stream>>>(maxkey, den, bufB, bias + l * DH, sums, N);
        // 4-6) edge passes: logits+max, exp+sum, normalize+scatter
        edge_logits_kernel<<<gE, TB, 0, stream>>>(ei, ea, asrc, adst, weA + l * F_E,
                                                  elog, maxkey, E);
        edge_exp_kernel<<<gE, TB, 0, stream>>>(ei, elog, maxkey, den, E);
        edge_aggr_kernel<<<gE32, TB, 0, stream>>>(ei, elog, den, bufA, bufB, E);
        // 7-8) BatchNorm + ReLU
        bn_stats_kernel<<<gN32, TB, 0, stream>>>(bufB, sums, N);
        bn_apply_kernel<<<gN32, TB, 0, stream>>>(bufB, sums, gamma + l * DH, beta + l * DH, N);
    }

    // graph pooling + head
    zero_kernel<<<(NB * DH + TB - 1) / TB, TB, 0, stream>>>(gemb, NB * DH);
    pool_kernel<<<gN32, TB, 0, stream>>>(bufB, batch, gemb, N);
    head_kernel<<<(NB + TB - 1) / TB, TB, 0, stream>>>(gemb, fcW, fcb, out, NB);
}